// MLoss_68066641707785
// MI455X (gfx1250) — compile-verified
//
#include <hip/hip_runtime.h>
#include <math.h>

typedef __attribute__((ext_vector_type(2))) float v2f;
typedef __attribute__((ext_vector_type(8))) float v8f;

#define THRESH 0.5f
#define ALPHA  0.5f

// Fold 32-lane partials (packed pairwise into B VGPR0/VGPR1) with one
// v_wmma_f32_16x16x4_f32 each: with A = ones on K in {0,2} (a02) the result
// D[m][n] = B[0][n] + B[2][n] = s.x[n] + s.x[n+16]; with A = ones on K in
// {1,3} (a13) it selects the .y component. Exact f32 adds on the matrix pipe.
__device__ __forceinline__ float wave_sum_wmma(float v, bool second, v2f packed) {
    v2f a;
    if (!second) { a[0] = 1.0f; a[1] = 0.0f; }
    else         { a[0] = 0.0f; a[1] = 1.0f; }
    v8f c = {};
    c = __builtin_amdgcn_wmma_f32_16x16x4_f32(false, a, false, packed,
                                              (short)0, c, false, false);
    float s = c[0];                 // lane l holds s[l%16] + s[l%16 + 16]
    s += __shfl_xor(s, 8, 32);
    s += __shfl_xor(s, 4, 32);
    s += __shfl_xor(s, 2, 32);
    s += __shfl_xor(s, 1, 32);
    (void)v;
    return s;                       // every lane: full 32-lane sum
}

__global__ __launch_bounds__(256)
void mloss_main(const float* __restrict__ x, const float* __restrict__ y,
                float* __restrict__ partials, long long groups) {
    const long long tid    = (long long)blockIdx.x * blockDim.x + threadIdx.x;
    const long long stride = (long long)gridDim.x * blockDim.x;

    float cnt = 0.0f, se = 0.0f, bce = 0.0f, bg = 0.0f;

    for (long long g = tid; g < groups; g += stride) {
        // prefetch next grid-stride chunk (global_prefetch_b8)
        long long gn = g + stride;
        if (gn < groups) {
            __builtin_prefetch(x + gn * 20, 0, 0);
            __builtin_prefetch(y + gn * 20, 0, 0);
        }
        // 4 rows x 5 channels = 20 floats = 5 aligned float4 loads per tensor
        const float4* xb = (const float4*)(x + g * 20);
        const float4* yb = (const float4*)(y + g * 20);
        float4 xv[5], yv[5];
#pragma unroll
        for (int i = 0; i < 5; ++i) { xv[i] = xb[i]; yv[i] = yb[i]; }
        const float* xf = (const float*)xv;
        const float* yf = (const float*)yv;
#pragma unroll
        for (int r = 0; r < 4; ++r) {
            float p = xf[5 * r + 0];
            float t = yf[5 * r + 0];
            float m = (t > THRESH) ? 1.0f : 0.0f;
            float lp  = logf(p);
            float l1m = log1pf(-p);
            float b = -(t * lp + (1.0f - t) * l1m);
            float d1 = xf[5 * r + 1] - yf[5 * r + 1];
            float d2 = xf[5 * r + 2] - yf[5 * r + 2];
            float d3 = xf[5 * r + 3] - yf[5 * r + 3];
            float d4 = xf[5 * r + 4] - yf[5 * r + 4];
            float s  = d1 * d1 + d2 * d2 + d3 * d3 + d4 * d4;
            cnt += m;
            se  += m * s;
            bce += m * b;
            bg  += -l1m;
        }
    }

    // ---- wave reduction via WMMA (all lanes converged here; EXEC == ones) ----
    v2f b0; b0[0] = se;  b0[1] = bce;
    v2f b1; b1[0] = cnt; b1[1] = bg;
    float se_w  = wave_sum_wmma(se,  false, b0);
    float bce_w = wave_sum_wmma(bce, true,  b0);
    float cnt_w = wave_sum_wmma(cnt, false, b1);
    float bg_w  = wave_sum_wmma(bg,  true,  b1);

    // ---- block reduction (8 waves), deterministic fixed order ----
    __shared__ float s_part[8][4];
    int lane = threadIdx.x & 31;
    int wave = threadIdx.x >> 5;
    if (lane == 0) {
        s_part[wave][0] = cnt_w;
        s_part[wave][1] = se_w;
        s_part[wave][2] = bce_w;
        s_part[wave][3] = bg_w;
    }
    __syncthreads();
    if (threadIdx.x == 0) {
        float a0 = 0, a1 = 0, a2 = 0, a3 = 0;
        int nw = blockDim.x >> 5;
        for (int w = 0; w < nw; ++w) {
            a0 += s_part[w][0]; a1 += s_part[w][1];
            a2 += s_part[w][2]; a3 += s_part[w][3];
        }
        float* out = partials + (long long)blockIdx.x * 4;
        out[0] = a0; out[1] = a1; out[2] = a2; out[3] = a3;
    }
}

__global__ __launch_bounds__(256)
void mloss_final(const float* __restrict__ partials, int nblk,
                 const float* __restrict__ x, const float* __restrict__ y,
                 long long tail_row, long long rows, float* __restrict__ out) {
    float a[4] = {0.0f, 0.0f, 0.0f, 0.0f};
    for (int i = threadIdx.x; i < nblk; i += blockDim.x) {
#pragma unroll
        for (int q = 0; q < 4; ++q) a[q] += partials[(long long)i * 4 + q];
    }
    // wave tree (fixed order)
#pragma unroll
    for (int q = 0; q < 4; ++q) {
        a[q] += __shfl_xor(a[q], 16, 32);
        a[q] += __shfl_xor(a[q], 8, 32);
        a[q] += __shfl_xor(a[q], 4, 32);
        a[q] += __shfl_xor(a[q], 2, 32);
        a[q] += __shfl_xor(a[q], 1, 32);
    }
    __shared__ float s_part[8][4];
    int lane = threadIdx.x & 31;
    int wave = threadIdx.x >> 5;
    if (lane == 0) {
#pragma unroll
        for (int q = 0; q < 4; ++q) s_part[wave][q] = a[q];
    }
    __syncthreads();
    if (threadIdx.x == 0) {
        float cnt = 0, se = 0, bce = 0, bg = 0;
        int nw = blockDim.x >> 5;
        for (int w = 0; w < nw; ++w) {
            cnt += s_part[w][0]; se += s_part[w][1];
            bce += s_part[w][2]; bg += s_part[w][3];
        }
        // tail rows not covered by 4-row groups (0 for the given shape)
        for (long long r = tail_row; r < rows; ++r) {
            float p = x[r * 5 + 0];
            float t = y[r * 5 + 0];
            float m = (t > THRESH) ? 1.0f : 0.0f;
            float lp = logf(p), l1m = log1pf(-p);
            float b = -(t * lp + (1.0f - t) * l1m);
            float d1 = x[r * 5 + 1] - y[r * 5 + 1];
            float d2 = x[r * 5 + 2] - y[r * 5 + 2];
            float d3 = x[r * 5 + 3] - y[r * 5 + 3];
            float d4 = x[r * 5 + 4] - y[r * 5 + 4];
            cnt += m;
            se  += m * (d1 * d1 + d2 * d2 + d3 * d3 + d4 * d4);
            bce += m * b;
            bg  += -l1m;
        }
        float face  = cnt;
        float scale = 1.0f + 1.0f / face;
        float diff_box = scale * se / (face * 4.0f);
        float diff_c   = scale * bce / face;
        float diff_bg  = ALPHA * bg / (float)rows;
        out[0] = diff_box + diff_c + diff_bg;
    }
}

extern "C" void kernel_launch(void* const* d_in, const int* in_sizes, int n_in,
                              void* d_out, int out_size, void* d_ws, size_t ws_size,
                              hipStream_t stream) {
    const float* x = (const float*)d_in[0];
    const float* y = (const float*)d_in[1];
    long long n      = in_sizes[0];
    long long rows   = n / 5;
    long long groups = rows / 4;

    int G = 1024;
    size_t need_per_block = 4 * sizeof(float);
    if (ws_size < (size_t)G * need_per_block) {
        G = (int)(ws_size / need_per_block);
        if (G < 1) G = 1;
    }
    float* partials = (float*)d_ws;

    mloss_main<<<G, 256, 0, stream>>>(x, y, partials, groups);
    mloss_final<<<1, 256, 0, stream>>>(partials, G, x, y, groups * 4, rows,
                                       (float*)d_out);
}